// TopoMLP_18184891531696
// MI455X (gfx1250) — compile-verified
//
#include <hip/hip_runtime.h>

typedef float v2f __attribute__((ext_vector_type(2)));
typedef float v8f __attribute__((ext_vector_type(8)));

#define NS   8192   // samples
#define DB   64     // branches
#define DIN  64     // input dim (== branches)
#define DH1  128    // hidden 1
#define DH2  64     // hidden 2

__device__ __forceinline__ float sigmoidf(float v) {
    return 1.0f / (1.0f + __expf(-v));
}

__device__ __forceinline__ v2f ld2(const float* p) {
    return *(const v2f*)p;   // 8B-aligned by construction (even float index)
}

// ---- binary-input predicate: flag = all(x==0 || x==1) ----
__global__ void init_flag_kernel(int* f) { *f = 1; }

__global__ void check_binary_kernel(const float* __restrict__ x, int n, int* f) {
    int i = blockIdx.x * blockDim.x + threadIdx.x;
    int stride = gridDim.x * blockDim.x;
    for (; i < n; i += stride) {
        float v = x[i];
        if (v != 0.0f && v != 1.0f) { *f = 0; return; }
    }
}

// One wave handles one (branch, 32-row sample tile) = two 16-row WMMA tiles,
// so every B-fragment load feeds TWO v_wmma_f32_16x16x4_f32 ops.
// 256 row-tiles per branch, 64 branches -> 16384 waves -> 8192 blocks of 64.
__global__ void __launch_bounds__(64)
topo_mlp_kernel(const float* __restrict__ x,  const float* __restrict__ W0,
                const float* __restrict__ b0, const float* __restrict__ W1,
                const float* __restrict__ b1, const float* __restrict__ W2,
                const float* __restrict__ b2, float* __restrict__ out,
                const int* __restrict__ flag_p)
{
    __shared__ float lds_act[2][32 * DH1];   // 16KB per wave, 32KB per WG

    const int lane   = threadIdx.x & 31;
    const int wav    = threadIdx.x >> 5;
    const int wid    = blockIdx.x * 2 + wav;
    const int branch = wid >> 8;             // / 256
    const int tile   = wid & 255;
    const int r0     = tile * 32;
    const int half   = lane >> 4;            // lane group (0: lanes 0-15, 1: 16-31)
    const int l15    = lane & 15;

    const int binflag = flag_p ? flag_p[0] : 0;

    // ---------- A fragments for layer 0: rows r0..r0+31, K=64 in 16 blocks of 4 ----------
    // A layout (16x4 f32): lane (M + 16h), VGPR j holds A[M][4kb + j + 2h] -> contiguous pair.
    const float* XrowA = x + (size_t)(r0 + l15) * DIN + 2 * half;
    const float* XrowB = XrowA + 16 * DIN;
    v2f a0A[16], a0B[16];
    #pragma unroll
    for (int kb = 0; kb < 16; ++kb) {
        a0A[kb] = ld2(XrowA + 4 * kb);
        a0B[kb] = ld2(XrowB + 4 * kb);
    }

    const float* W0b = W0 + (size_t)branch * DIN * DH1;   // [d=64][h=128], h contiguous
    const float* b0b = b0 + branch * DH1;
    float* acts = lds_act[wav];                           // [32 rows][128 cols]

    // ---------- layer 0: acc[32x128] = X[32x64] * W0[64x128] + b0 ----------
    #pragma unroll
    for (int t = 0; t < 8; ++t) {
        const float bias = b0b[16 * t + l15];
        v8f accA = {bias, bias, bias, bias, bias, bias, bias, bias};
        v8f accB = accA;
        #pragma unroll
        for (int kb = 0; kb < 16; ++kb) {
            // B layout (4x16 f32): lane (N + 16h), VGPR j holds B[4kb + j + 2h][N]
            const float* bp = W0b + (size_t)(4 * kb + 2 * half) * DH1 + 16 * t + l15;
            v2f bf = {bp[0], bp[DH1]};     // adjacent K rows -> stride DH1
            accA = __builtin_amdgcn_wmma_f32_16x16x4_f32(
                false, a0A[kb], false, bf, (short)0, accA, false, false);
            accB = __builtin_amdgcn_wmma_f32_16x16x4_f32(
                false, a0B[kb], false, bf, (short)0, accB, false, false);
        }
        // sigmoid, stash row-major [row][128] in this wave's LDS slice.
        // C/D layout: VGPR j, lane L -> row (j + 8h), col (16t + l15)
        #pragma unroll
        for (int j = 0; j < 8; ++j) {
            acts[(j + 8 * half) * DH1 + 16 * t + l15]        = sigmoidf(accA[j]);
            acts[(16 + j + 8 * half) * DH1 + 16 * t + l15]   = sigmoidf(accB[j]);
        }
    }

    // same-wave DS ops are in-order; wait + memory clobber pins compiler ordering
    asm volatile("s_wait_dscnt 0" ::: "memory");

    // ---------- A fragments for layer 1: act tile [32 x 128], 32 K-blocks ----------
    v2f a1A[32], a1B[32];
    #pragma unroll
    for (int kb = 0; kb < 32; ++kb) {
        a1A[kb] = ld2(&acts[l15 * DH1 + 4 * kb + 2 * half]);
        a1B[kb] = ld2(&acts[(16 + l15) * DH1 + 4 * kb + 2 * half]);
    }

    const float* W1b = W1 + (size_t)branch * DH2 * DH1;   // [o=64][h=128], h contiguous
    const float* b1b = b1 + branch * DH2;
    const float* W2b = W2 + branch * DH2;                 // [o=64]

    float pA[8], pB[8];
    #pragma unroll
    for (int j = 0; j < 8; ++j) { pA[j] = 0.0f; pB[j] = 0.0f; }

    // ---------- layer 1 (+ fused layer-2 dot): acc[32x64] = act * W1^T + b1 ----------
    #pragma unroll
    for (int t = 0; t < 4; ++t) {
        const float bias = b1b[16 * t + l15];
        v8f accA = {bias, bias, bias, bias, bias, bias, bias, bias};
        v8f accB = accA;
        #pragma unroll
        for (int kb = 0; kb < 32; ++kb) {
            // B[K=h][N=o] = W1[i][o][h]: the j=0,1 pair is contiguous in h
            const float* bp = W1b + (size_t)(16 * t + l15) * DH1 + 4 * kb + 2 * half;
            v2f bf = ld2(bp);
            accA = __builtin_amdgcn_wmma_f32_16x16x4_f32(
                false, a1A[kb], false, bf, (short)0, accA, false, false);
            accB = __builtin_amdgcn_wmma_f32_16x16x4_f32(
                false, a1B[kb], false, bf, (short)0, accB, false, false);
        }
        const float w2v = W2b[16 * t + l15];
        #pragma unroll
        for (int j = 0; j < 8; ++j) {
            pA[j] += sigmoidf(accA[j]) * w2v;   // layer-2 contribution of col (16t+l15)
            pB[j] += sigmoidf(accB[j]) * w2v;
        }
    }

    // ---------- layer-2 reduction across the 16 lanes of each half ----------
    const float bias2 = b2[branch];
    #pragma unroll
    for (int j = 0; j < 8; ++j) {
        float vA = pA[j];
        vA += __shfl_xor(vA, 1, 32);
        vA += __shfl_xor(vA, 2, 32);
        vA += __shfl_xor(vA, 4, 32);
        vA += __shfl_xor(vA, 8, 32);    // offsets <16 stay within each half
        float vB = pB[j];
        vB += __shfl_xor(vB, 1, 32);
        vB += __shfl_xor(vB, 2, 32);
        vB += __shfl_xor(vB, 4, 32);
        vB += __shfl_xor(vB, 8, 32);
        if (l15 == 0) {
            const int rowA = r0 + j + 8 * half;
            float oA = vA + bias2;
            float oB = vB + bias2;
            if (binflag) { oA = sigmoidf(oA); oB = sigmoidf(oB); }
            out[(size_t)rowA * DB + branch]        = oA;
            out[(size_t)(rowA + 16) * DB + branch] = oB;
        }
    }
}

extern "C" void kernel_launch(void* const* d_in, const int* in_sizes, int n_in,
                              void* d_out, int out_size, void* d_ws, size_t ws_size,
                              hipStream_t stream) {
    const float* x  = (const float*)d_in[0];
    const float* W0 = (const float*)d_in[1];
    const float* b0 = (const float*)d_in[2];
    const float* W1 = (const float*)d_in[3];
    const float* b1 = (const float*)d_in[4];
    const float* W2 = (const float*)d_in[5];
    const float* b2 = (const float*)d_in[6];
    float* out = (float*)d_out;

    int* flag = (ws_size >= sizeof(int)) ? (int*)d_ws : nullptr;
    if (flag) {
        init_flag_kernel<<<1, 1, 0, stream>>>(flag);
        check_binary_kernel<<<256, 256, 0, stream>>>(x, NS * DIN, flag);
    }

    const int total_waves = DB * (NS / 32);       // 16384
    dim3 grid(total_waves / 2);                   // 2 waves (64 thr) per block
    topo_mlp_kernel<<<grid, 64, 0, stream>>>(x, W0, b0, W1, b1, W2, b2, out, flag);
}